// GAT_89842125897777
// MI455X (gfx1250) — compile-verified
//
#include <hip/hip_runtime.h>
#include <math.h>

#define N_NODES 50000
#define N_EDGES 800000
#define E_TOT   (N_EDGES + N_NODES)   // self loops appended
#define F_IN    256
#define HEADS   8
#define OUT_CH  10
#define HIDDEN  80                    // HEADS*OUT_CH

typedef __attribute__((ext_vector_type(2))) float v2f;
typedef __attribute__((ext_vector_type(8))) float v8f;

// float atomic max via signed/unsigned integer ordering trick
__device__ __forceinline__ void atomic_max_f32(float* addr, float val) {
  if (val >= 0.0f) atomicMax((int*)addr, __float_as_int(val));
  else             atomicMin((unsigned int*)addr, __float_as_uint(val));
}

// ---------------------------------------------------------------------------
// GEMM: xl = X * Wl, xr = X * Wr     X:[n,K]  W:[K,80]  out:[n,80]
// One wave per 16x16 output tile; fp32 WMMA 16x16x4; A fragment reused for
// both weight matrices. Tiling is exact (n=3125*16, 80=5*16, K%4==0) so EXEC
// stays all-ones around every WMMA.
// A frag (16x4 f32): lanes 0-15 -> K=k,k+1 ; lanes 16-31 -> K=k+2,k+3
// B frag (4x16 f32): VGPR0 = rows K=k (lo lanes) / k+2 (hi), VGPR1 = k+1 / k+3
// C/D: VGPR v, lanes 0-15 -> M=v, lanes 16-31 -> M=v+8, N=lane%16
// ---------------------------------------------------------------------------
__global__ void __launch_bounds__(128)
gat_gemm2_wmma(const float* __restrict__ X,
               const float* __restrict__ Wl,
               const float* __restrict__ Wr,
               float* __restrict__ xl,
               float* __restrict__ xr,
               int K) {
  const int NT = HIDDEN / 16;                       // 5 N-tiles
  const int n_tiles = (N_NODES / 16) * NT;          // 15625 tiles
  int wave = blockIdx.x * (blockDim.x >> 5) + (threadIdx.x >> 5);
  if (wave >= n_tiles) return;                      // wave-uniform exit

  int mt = wave / NT;
  int nt = wave % NT;
  int lane = threadIdx.x & 31;
  int half = lane >> 4;                             // 0: K pair 0/1, 1: K pair 2/3
  int r    = lane & 15;

  int arow = mt * 16 + r;                           // A row handled by this lane
  int bcol = nt * 16 + r;                           // B/C column handled by this lane

  v8f cl = {};
  v8f cr = {};

  for (int k = 0; k < K; k += 4) {
    int ka = k + 2 * half;
    __builtin_prefetch(&X[arow * K + k + 16], 0, 0);
    v2f a;
    a.x = X[arow * K + ka];
    a.y = X[arow * K + ka + 1];
    v2f bl;
    bl.x = Wl[ka * HIDDEN + bcol];
    bl.y = Wl[(ka + 1) * HIDDEN + bcol];
    v2f br;
    br.x = Wr[ka * HIDDEN + bcol];
    br.y = Wr[(ka + 1) * HIDDEN + bcol];
    cl = __builtin_amdgcn_wmma_f32_16x16x4_f32(false, a, false, bl, (short)0, cl, false, false);
    cr = __builtin_amdgcn_wmma_f32_16x16x4_f32(false, a, false, br, (short)0, cr, false, false);
  }

#pragma unroll
  for (int v = 0; v < 8; ++v) {
    int orow = mt * 16 + v + 8 * half;
    xl[orow * HIDDEN + nt * 16 + r] = cl[v];
    xr[orow * HIDDEN + nt * 16 + r] = cr[v];
  }
}

// ---------------------------------------------------------------------------
// Init: m = -inf, denom = 0, acc = 0
// ---------------------------------------------------------------------------
__global__ void gat_init(float* __restrict__ m, float* __restrict__ denom,
                         float* __restrict__ acc) {
  int idx = blockIdx.x * blockDim.x + threadIdx.x;
  if (idx < N_NODES * HEADS) {
    m[idx]     = -__builtin_inff();
    denom[idx] = 0.0f;
  }
  if (idx < N_NODES * HIDDEN) acc[idx] = 0.0f;
}

// ---------------------------------------------------------------------------
// Pass 1: per (edge, head) attention logit + segment max into m[dst,h]
// ---------------------------------------------------------------------------
__global__ void gat_edge_logits(const int* __restrict__ esrc,
                                const int* __restrict__ edst,
                                const float* __restrict__ xl,
                                const float* __restrict__ xr,
                                const float* __restrict__ att,
                                float* __restrict__ logit,
                                float* m_buf) {
  int eh = blockIdx.x * blockDim.x + threadIdx.x;
  if (eh >= E_TOT * HEADS) return;
  int e = eh >> 3;
  int h = eh & 7;
  int s, d;
  if (e < N_EDGES) { s = esrc[e]; d = edst[e]; }
  else             { s = d = e - N_EDGES; }        // self loop

  const float* pl = xl + s * HIDDEN + h * OUT_CH;
  const float* pr = xr + d * HIDDEN + h * OUT_CH;
  const float* pa = att + h * OUT_CH;
  float acc = 0.0f;
#pragma unroll
  for (int c = 0; c < OUT_CH; ++c) {
    float v = pl[c] + pr[c];
    v = v > 0.0f ? v : 0.2f * v;                   // LeakyReLU(0.2)
    acc += v * pa[c];
  }
  logit[eh] = acc;
  atomic_max_f32(&m_buf[d * HEADS + h], acc);
}

// ---------------------------------------------------------------------------
// Pass 2: ex = exp(logit - m[dst]); denom[dst,h] += ex;
//         acc[dst, h*10+c] += ex * xl[src, h*10+c]   (unnormalized)
// ---------------------------------------------------------------------------
__global__ void gat_edge_accum(const int* __restrict__ esrc,
                               const int* __restrict__ edst,
                               const float* __restrict__ xl,
                               const float* __restrict__ logit,
                               const float* __restrict__ m_buf,
                               float* denom,
                               float* acc) {
  int eh = blockIdx.x * blockDim.x + threadIdx.x;
  if (eh >= E_TOT * HEADS) return;
  int e = eh >> 3;
  int h = eh & 7;
  int s, d;
  if (e < N_EDGES) { s = esrc[e]; d = edst[e]; }
  else             { s = d = e - N_EDGES; }

  float ex = __expf(logit[eh] - m_buf[d * HEADS + h]);
  atomicAdd(&denom[d * HEADS + h], ex);
  const float* pl = xl + s * HIDDEN + h * OUT_CH;
  float* pa = acc + d * HIDDEN + h * OUT_CH;
#pragma unroll
  for (int c = 0; c < OUT_CH; ++c) atomicAdd(&pa[c], ex * pl[c]);
}

// ---------------------------------------------------------------------------
// Node finish: out = elu(acc/(denom+1e-16) + bias)
// ---------------------------------------------------------------------------
__global__ void gat_node_finish(const float* __restrict__ acc,
                                const float* __restrict__ denom,
                                const float* __restrict__ bias,
                                float* __restrict__ xout) {
  int idx = blockIdx.x * blockDim.x + threadIdx.x;
  if (idx >= N_NODES * HIDDEN) return;
  int i = idx / HIDDEN;
  int j = idx - i * HIDDEN;
  int h = j / OUT_CH;
  float v = acc[idx] / (denom[i * HEADS + h] + 1e-16f) + bias[j];
  xout[idx] = v > 0.0f ? v : expm1f(v);            // ELU(alpha=1)
}

// ---------------------------------------------------------------------------
extern "C" void kernel_launch(void* const* d_in, const int* in_sizes, int n_in,
                              void* d_out, int out_size, void* d_ws, size_t ws_size,
                              hipStream_t stream) {
  (void)in_sizes; (void)n_in; (void)out_size; (void)ws_size;

  const float* x    = (const float*)d_in[0];
  const int*   ei   = (const int*)d_in[1];         // [2, N_EDGES] flat (int32 under jax default x32)
  const int*   esrc = ei;
  const int*   edst = ei + N_EDGES;

  // Workspace layout (floats)
  float* ws    = (float*)d_ws;
  float* xl    = ws;                               // 4,000,000
  float* xr    = xl    + (size_t)N_NODES * HIDDEN; // 4,000,000
  float* logit = xr    + (size_t)N_NODES * HIDDEN; // 6,800,000
  float* mbuf  = logit + (size_t)E_TOT * HEADS;    //   400,000
  float* denom = mbuf  + (size_t)N_NODES * HEADS;  //   400,000
  float* acc   = denom + (size_t)N_NODES * HEADS;  // 4,000,000
  float* xbuf  = acc   + (size_t)N_NODES * HIDDEN; // 4,000,000
  // total ~23.6M floats (~94.4 MB)

  const int gemm_waves  = (N_NODES / 16) * (HIDDEN / 16);   // 15625
  const int gemm_blocks = (gemm_waves + 3) / 4;             // 4 waves / block
  const int node_blocks = (N_NODES * HIDDEN + 255) / 256;
  const int edge_blocks = (E_TOT * HEADS + 255) / 256;

  const float* xin = x;
  int K = F_IN;
  for (int L = 0; L < 3; ++L) {
    const float* Wl = (const float*)d_in[2 + 4 * L];
    const float* Wr = (const float*)d_in[3 + 4 * L];
    const float* at = (const float*)d_in[4 + 4 * L];
    const float* b  = (const float*)d_in[5 + 4 * L];

    gat_gemm2_wmma<<<gemm_blocks, 128, 0, stream>>>(xin, Wl, Wr, xl, xr, K);
    gat_init<<<node_blocks, 256, 0, stream>>>(mbuf, denom, acc);
    gat_edge_logits<<<edge_blocks, 256, 0, stream>>>(esrc, edst, xl, xr, at, logit, mbuf);
    gat_edge_accum<<<edge_blocks, 256, 0, stream>>>(esrc, edst, xl, logit, mbuf, denom, acc);

    float* xout = (L == 2) ? (float*)d_out : xbuf;
    gat_node_finish<<<node_blocks, 256, 0, stream>>>(acc, denom, b, xout);

    xin = xbuf;
    K = HIDDEN;
  }
}